// SamplingLayer_19911468385054
// MI455X (gfx1250) — compile-verified
//
#include <hip/hip_runtime.h>
#include <hip/hip_bf16.h>

#define NUM_SAMPLED 8192
#define NUM_CLASSES 50000
#define DIM 512
#define BATCH 8192

#define BM 128
#define BN 128
#define KC 32
#define LDA 36   // padded LDS row stride in floats (16B-aligned rows, bank-conflict free)

typedef float v2f __attribute__((ext_vector_type(2)));
typedef float v8f __attribute__((ext_vector_type(8)));

// expected-count term of the log-uniform sampler
__device__ __forceinline__ float log_uniform_expected(int c) {
    float cf = (float)c;
    float p = (logf(cf + 2.0f) - logf(cf + 1.0f)) * (1.0f / logf((float)(NUM_CLASSES + 1)));
    return -expm1f((float)NUM_SAMPLED * log1pf(-p));
}

// K1: per-sampled-column additive term, and zero the row accumulators
__global__ __launch_bounds__(256) void prep_kernel(const float* __restrict__ bias,
                                                   const int* __restrict__ sampled,
                                                   float* __restrict__ adj,
                                                   float* __restrict__ acc) {
    int i = blockIdx.x * 256 + threadIdx.x;
    if (i < NUM_SAMPLED) {
        int s = sampled[i];
        adj[i] = bias[s] - logf(log_uniform_expected(s));
        acc[i] = 0.0f;   // NUM_SAMPLED == BATCH, so this zeroes all row accumulators
    }
}

// K2: true logits, one wave per batch row
__global__ __launch_bounds__(256) void true_logits_kernel(const float* __restrict__ X,
                                                          const float* __restrict__ W,
                                                          const float* __restrict__ bias,
                                                          const int* __restrict__ labels,
                                                          float* __restrict__ tl) {
    int wv   = threadIdx.x >> 5;
    int lane = threadIdx.x & 31;
    int b = blockIdx.x * 8 + wv;
    if (b >= BATCH) return;
    int L = labels[b];
    const float* x = X + (size_t)b * DIM;
    const float* w = W + (size_t)L * DIM;
    float s = 0.0f;
#pragma unroll
    for (int j = 0; j < 4; j++) {
        int d4 = (lane + 32 * j) * 4;
        float4 xv  = *(const float4*)(x + d4);
        float4 wv4 = *(const float4*)(w + d4);
        s += xv.x * wv4.x + xv.y * wv4.y + xv.z * wv4.z + xv.w * wv4.w;
    }
#pragma unroll
    for (int m = 1; m < 32; m <<= 1) s += __shfl_xor(s, m, 32);
    if (lane == 0)
        tl[b] = s + bias[L] - logf(log_uniform_expected(L));
}

// K3: fused GEMM (inputs @ W[sampled]^T) + bias/correction + hit-mask + per-row exp-sum
// Block tile 128x128 (8 waves, each 32 rows x 64 cols = 8 f32 WMMA accumulators)
__global__ __launch_bounds__(256) void gemm_expsum_kernel(const float* __restrict__ X,
                                                          const float* __restrict__ W,
                                                          const int* __restrict__ labels,
                                                          const int* __restrict__ sampled,
                                                          const float* __restrict__ adj,
                                                          float* __restrict__ acc) {
    __shared__ float As[BM][LDA];
    __shared__ float Bs[BN][LDA];
    __shared__ int   sid[BN];
    __shared__ int   lab[BM];
    __shared__ float badj[BN];

    const int tid     = threadIdx.x;
    const int rowbase = blockIdx.y * BM;
    const int colbase = blockIdx.x * BN;

    if (tid < BN) {
        sid[tid]  = sampled[colbase + tid];
        badj[tid] = adj[colbase + tid];
    } else {
        lab[tid - BN] = labels[rowbase + (tid - BN)];
    }
    __syncthreads();

    const int lane = tid & 31;
    const int wv   = tid >> 5;
    const int wrow = (wv & 3) * 32;    // 0,32,64,96
    const int wcol = (wv >> 2) * 64;   // 0,64

    v8f c[2][4];
#pragma unroll
    for (int r = 0; r < 2; r++)
#pragma unroll
        for (int j = 0; j < 4; j++) c[r][j] = (v8f){};

    // global->LDS copy coordinates: 256 threads move 128x32 floats per matrix per chunk
    const int lrow = tid >> 3;         // 0..31
    const int lseg = (tid & 7) * 4;    // 0..28

    const int m  = lane & 15;
    const int kh = (lane >> 4) * 2;    // lanes 0-15: K+0..1, lanes 16-31: K+2..3

    for (int k0 = 0; k0 < DIM; k0 += KC) {
        float4 ar[4], br[4];
#pragma unroll
        for (int r = 0; r < 4; r++) {
            ar[r] = *(const float4*)&X[(size_t)(rowbase + lrow + 32 * r) * DIM + k0 + lseg];
            br[r] = *(const float4*)&W[(size_t)sid[lrow + 32 * r] * DIM + k0 + lseg];
        }
        __syncthreads();   // previous iteration's LDS reads complete
#pragma unroll
        for (int r = 0; r < 4; r++) {
            *(float4*)&As[lrow + 32 * r][lseg] = ar[r];
            *(float4*)&Bs[lrow + 32 * r][lseg] = br[r];
        }
        __syncthreads();

#pragma unroll
        for (int kk = 0; kk < KC; kk += 4) {
            v2f a0 = *(const v2f*)&As[wrow + m][kk + kh];
            v2f a1 = *(const v2f*)&As[wrow + 16 + m][kk + kh];
            v2f b0 = *(const v2f*)&Bs[wcol + m][kk + kh];
            v2f b1 = *(const v2f*)&Bs[wcol + 16 + m][kk + kh];
            v2f b2 = *(const v2f*)&Bs[wcol + 32 + m][kk + kh];
            v2f b3 = *(const v2f*)&Bs[wcol + 48 + m][kk + kh];
            c[0][0] = __builtin_amdgcn_wmma_f32_16x16x4_f32(false, a0, false, b0, (short)0, c[0][0], false, false);
            c[0][1] = __builtin_amdgcn_wmma_f32_16x16x4_f32(false, a0, false, b1, (short)0, c[0][1], false, false);
            c[0][2] = __builtin_amdgcn_wmma_f32_16x16x4_f32(false, a0, false, b2, (short)0, c[0][2], false, false);
            c[0][3] = __builtin_amdgcn_wmma_f32_16x16x4_f32(false, a0, false, b3, (short)0, c[0][3], false, false);
            c[1][0] = __builtin_amdgcn_wmma_f32_16x16x4_f32(false, a1, false, b0, (short)0, c[1][0], false, false);
            c[1][1] = __builtin_amdgcn_wmma_f32_16x16x4_f32(false, a1, false, b1, (short)0, c[1][1], false, false);
            c[1][2] = __builtin_amdgcn_wmma_f32_16x16x4_f32(false, a1, false, b2, (short)0, c[1][2], false, false);
            c[1][3] = __builtin_amdgcn_wmma_f32_16x16x4_f32(false, a1, false, b3, (short)0, c[1][3], false, false);
        }
    }

    // Epilogue: C layout: VGPR v -> row (v + 8*(lane>>4)), col = lane&15 within 16-col tile
    const int n    = lane & 15;
    const int half = lane >> 4;
    int   sc[4];
    float jc[4];
#pragma unroll
    for (int j = 0; j < 4; j++) {
        sc[j] = sid[wcol + 16 * j + n];
        jc[j] = badj[wcol + 16 * j + n];
    }
#pragma unroll
    for (int r = 0; r < 2; r++) {
        const int rowb = wrow + 16 * r;
#pragma unroll
        for (int v = 0; v < 8; v++) {
            int mrow = v + 8 * half;
            int L    = lab[rowb + mrow];
            float e = 0.0f;
#pragma unroll
            for (int j = 0; j < 4; j++) {
                float l = c[r][j][v] + jc[j];
                e += (L == sc[j]) ? 0.0f : __expf(l);
            }
#pragma unroll
            for (int msk = 1; msk < 16; msk <<= 1) e += __shfl_xor(e, msk, 32);
            if (n == 0)
                atomicAdd(&acc[rowbase + rowb + mrow], e);
        }
    }
}

// K4: loss = log(sum_exp_sampled + exp(true)) - true
__global__ __launch_bounds__(256) void finalize_kernel(const float* __restrict__ acc,
                                                       const float* __restrict__ tl,
                                                       float* __restrict__ loss) {
    int b = blockIdx.x * 256 + threadIdx.x;
    if (b < BATCH) {
        float t = tl[b];
        loss[b] = logf(acc[b] + __expf(t)) - t;
    }
}

extern "C" void kernel_launch(void* const* d_in, const int* in_sizes, int n_in,
                              void* d_out, int out_size, void* d_ws, size_t ws_size,
                              hipStream_t stream) {
    const float* inputs  = (const float*)d_in[0];   // [8192][512]
    const int*   labels  = (const int*)d_in[1];     // [8192]
    const float* kernel  = (const float*)d_in[2];   // [50000][512]
    const float* bias    = (const float*)d_in[3];   // [50000]
    const int*   sampled = (const int*)d_in[4];     // [8192]
    float*       loss    = (float*)d_out;           // [8192]

    float* adj = (float*)d_ws;          // [8192]
    float* tl  = adj + NUM_SAMPLED;     // [8192]
    float* acc = tl + BATCH;            // [8192]

    prep_kernel<<<NUM_SAMPLED / 256, 256, 0, stream>>>(bias, sampled, adj, acc);
    true_logits_kernel<<<BATCH / 8, 256, 0, stream>>>(inputs, kernel, bias, labels, tl);
    dim3 grid(NUM_SAMPLED / BN, BATCH / BM);
    gemm_expsum_kernel<<<grid, 256, 0, stream>>>(inputs, kernel, labels, sampled, adj, acc);
    finalize_kernel<<<BATCH / 256, 256, 0, stream>>>(acc, tl, loss);
}